// AddNoise_31679678775522
// MI455X (gfx1250) — compile-verified
//
#include <hip/hip_runtime.h>
#include <hip/hip_bf16.h>

typedef __attribute__((ext_vector_type(2))) float v2f;
typedef __attribute__((ext_vector_type(8))) float v8f;

#define BB 2
#define TT 171
#define CC 512
#define HH 24
#define WW 24
#define HW 576
#define TGTN 384
#define KPATCH 4608   /* 9*512 */
#define NG 896        /* 512+256+128 */

#define O_IMG  0
#define O_TXT  589824
#define O_V0   764928
#define O_V1   1354752
#define O_V2   1649664
#define O_LOSS 1797120

// =====================================================================================
// Tiled wave32 WMMA f32 GEMM:  C[M,N] = epi(A[M,K] * B[K,N])
//   A row-major (lda=K), B row-major (ldb=N), C row-major (ldc) + optional z-batching.
//   Block = 256 threads = 8 waves; macro-tile 32(M) x 64(N); 2x4 wave grid of 16x16
//   WMMA tiles; K staged through LDS in 32-wide chunks with B128 loads/stores.
//   LDS padding: As stride 36, Bs stride 68 -> conflict-free frag reads, 16B-aligned
//   staging stores.
// =====================================================================================
__global__ void __launch_bounds__(256)
k_wmma_gemm_tiled(const float* __restrict__ A, int lda, long aZ,
                  const float* __restrict__ Bm, int ldb, long bZ,
                  const float* __restrict__ bias,
                  const float* __restrict__ rowScale, int rowScaleDiv,
                  float* __restrict__ C, int ldc, long cZ,
                  int M, int N, int K, int doRelu)
{
  __shared__ float As[32 * 36];
  __shared__ float Bs[32 * 68];

  int z = blockIdx.z;
  A  += (long)z * aZ;
  Bm += (long)z * bZ;
  C  += (long)z * cZ;

  int t    = threadIdx.x;
  int lane = t & 31;
  int l    = lane & 15;
  int hi   = lane >> 4;
  int wave = t >> 5;
  int wm   = wave >> 2;   // 0..1
  int wn   = wave & 3;    // 0..3
  int nBase = blockIdx.x * 64;
  int mBase = blockIdx.y * 32;

  // staging assignment: A tile 32x32 -> 1 float4/thread; B tile 32x64 -> 2 float4/thread
  int am = t >> 3, ak = (t & 7) << 2;
  int arow = mBase + am; if (arow > M - 1) arow = M - 1;       // clamp, no divergence
  const float* Ag = A + (size_t)arow * lda + ak;
  int bk = t >> 4, bn = (t & 15) << 2;
  const float* Bg = Bm + (size_t)bk * ldb + nBase + bn;
  size_t bstep16 = (size_t)16 * ldb;

  v8f acc = {};
  for (int k0 = 0; k0 < K; k0 += 32) {
    float4 a4  = *(const float4*)(Ag + k0);
    float4 b4a = *(const float4*)(Bg + (size_t)k0 * ldb);
    float4 b4b = *(const float4*)(Bg + (size_t)k0 * ldb + bstep16);
    if (k0 + 32 < K) {
      __builtin_prefetch(Ag + k0 + 32, 0, 0);
      __builtin_prefetch(Bg + (size_t)(k0 + 32) * ldb, 0, 0);
      __builtin_prefetch(Bg + (size_t)(k0 + 32) * ldb + bstep16, 0, 0);
    }
    __syncthreads();
    *(float4*)&As[am * 36 + ak]        = a4;
    *(float4*)&Bs[bk * 68 + bn]        = b4a;
    *(float4*)&Bs[(bk + 16) * 68 + bn] = b4b;
    __syncthreads();
#pragma unroll
    for (int kk = 0; kk < 32; kk += 4) {
      int ka = kk + (hi << 1);
      float2 af = *(const float2*)&As[(wm * 16 + l) * 36 + ka];   // ds_load_b64
      v2f av; av.x = af.x; av.y = af.y;
      v2f bv;
      bv.x = Bs[ka * 68 + wn * 16 + l];
      bv.y = Bs[(ka + 1) * 68 + wn * 16 + l];
      acc = __builtin_amdgcn_wmma_f32_16x16x4_f32(false, av, false, bv, (short)0, acc,
                                                  false, false);
    }
  }

  int n = nBase + wn * 16 + l;
  float bb = bias ? bias[n] : 0.0f;
#pragma unroll
  for (int r = 0; r < 8; ++r) {
    int m = mBase + wm * 16 + r + (hi << 3);
    if (m < M) {
      float v = acc[r] + bb;
      if (doRelu) v = fmaxf(v, 0.0f);
      if (rowScale) v *= rowScale[m / rowScaleDiv];
      C[(size_t)m * ldc + n] = v;
    }
  }
}

// ---------------- conv1 tap-sum matrices, TRANSPOSED: Wsum[cat][c][o] ([k][n] row-major) ----
__global__ void k_wsum(const float* __restrict__ w1, float* __restrict__ Wsum) {
  int idx = blockIdx.x * blockDim.x + threadIdx.x;
  if (idx >= 9 * CC * CC) return;
  int cat = idx / (CC * CC);
  int rem = idx - cat * CC * CC;
  int c = rem / CC, o = rem % CC;
  int rc = cat / 3, cc = cat % 3;
  int ky0 = (rc == 0) ? 1 : 0, ky1 = (rc == 2) ? 1 : 2;
  int kx0 = (cc == 0) ? 1 : 0, kx1 = (cc == 2) ? 1 : 2;
  const float* base = w1 + (size_t)(o * CC + c) * 9;
  float s = 0.0f;
  for (int ky = ky0; ky <= ky1; ++ky)
    for (int kx = kx0; kx <= kx1; ++kx)
      s += base[ky * 3 + kx];
  Wsum[idx] = s;
}

// ---------------- conv2_w reshaped to [k=(tap*512+c)][n=o] ----------------
__global__ void k_w2c(const float* __restrict__ w2, float* __restrict__ W2c) {
  int idx = blockIdx.x * blockDim.x + threadIdx.x;
  if (idx >= KPATCH * CC) return;
  int k = idx / CC, n = idx % CC;
  int tap = k / CC, c = k % CC;
  W2c[idx] = w2[((size_t)(n * CC + c) * 9) + tap];
}

// ---------------- g0|g1|g2 weights combined to [k][n=0..895] + combined bias ----------------
__global__ void k_gc(const float* __restrict__ g0w, const float* __restrict__ g1w,
                     const float* __restrict__ g2w, const float* __restrict__ g0b,
                     const float* __restrict__ g1b, const float* __restrict__ g2b,
                     float* __restrict__ Gc, float* __restrict__ gbias) {
  long idx = (long)blockIdx.x * blockDim.x + threadIdx.x;
  if (idx >= (long)KPATCH * NG) return;
  int n = (int)(idx % NG);
  int k = (int)(idx / NG);
  int tap = k / CC, c = k % CC;
  float v;
  if (n < 512)      v = g0w[((size_t)(n * CC + c) * 9) + tap];
  else if (n < 768) v = g1w[((size_t)((n - 512) * CC + c) * 9) + tap];
  else              v = g2w[((size_t)((n - 768) * CC + c) * 9) + tap];
  Gc[idx] = v;
  if (k == 0) gbias[n] = (n < 512) ? g0b[n] : (n < 768) ? g1b[n - 512] : g2b[n - 768];
}

// ---------------- per-batch distinct-value counts ----------------
__global__ void k_count(const int* __restrict__ targets, int* __restrict__ cnts) {
  __shared__ int bins[256];
  __shared__ int pres[TT];
  int b = blockIdx.x, tid = threadIdx.x;
  for (int i = tid; i < 256; i += blockDim.x) bins[i] = 0;
  for (int i = tid; i < TT; i += blockDim.x) pres[i] = 0;
  __syncthreads();
  const int* tb = targets + (size_t)b * TGTN * TGTN;
  for (int i = tid; i < TGTN * TGTN; i += blockDim.x) {
    int v = tb[i];
    if (v >= 0 && v < 256) atomicAdd(&bins[v], 1);
  }
  for (int i = tid; i < HW; i += blockDim.x) {
    int h = i / WW, w = i % WW;
    int v = tb[h * 16 * TGTN + w * 16];
    if (v >= 0 && v < TT) pres[v] = 1;
  }
  __syncthreads();
  if (tid == 0) {
    int cf = 0;
    for (int v = 0; v < 255; ++v) cf += (bins[v] > 0);
    int cd = 0;
    for (int v = 0; v < TT; ++v) cd += pres[v];
    cnts[b * 2 + 0] = cd;
    cnts[b * 2 + 1] = cf;
  }
}

// ---------------- nan / deficit / rank / extra gate (B=2) + loss ----------------
__global__ void k_gate(const int* __restrict__ cnts, float* __restrict__ gate,
                       float* __restrict__ loss_out) {
  bool nan0 = (cnts[0] != cnts[1]) || (cnts[1] == 0);
  bool nan1 = (cnts[2] != cnts[3]) || (cnts[3] == 0);
  int deficit = 1 - ((nan0 ? 1 : 0) + (nan1 ? 1 : 0));
  if (deficit < 0) deficit = 0;
  int r0 = nan0 ? 0 : 1;
  int r1 = r0 + (nan1 ? 0 : 1);
  bool ex0 = !nan0 && (r0 <= deficit);
  bool ex1 = !nan1 && (r1 <= deficit);
  gate[0] = (nan0 || ex0) ? 0.0f : 1.0f;
  gate[1] = (nan1 || ex1) ? 0.0f : 1.0f;
  *loss_out = 0.0f;
}

// rowGate[row] = gate[b] * (t valid), row = b*576+pos
__global__ void k_rowgate(const int* __restrict__ targets, const float* __restrict__ gate,
                          float* __restrict__ rowGate) {
  int row = blockIdx.x * blockDim.x + threadIdx.x;
  if (row >= BB * HW) return;
  int b = row / HW, pos = row % HW;
  int h = pos / WW, w = pos % WW;
  int t = targets[(size_t)b * TGTN * TGTN + h * 16 * TGTN + w * 16];
  rowGate[row] = (t >= 0 && t < TT) ? gate[b] : 0.0f;
}

__device__ __forceinline__ int rcat12(int i) { return (i == 0) ? 0 : ((i == 11) ? 2 : 1); }

// ---------------- bilinear-upsampled conv2 im2col: yup[row][tap][c] ----------------
__global__ void k_yup(const int* __restrict__ targets, const float* __restrict__ Y1,
                      float* __restrict__ yup) {
  long idx = (long)blockIdx.x * blockDim.x + threadIdx.x;
  if (idx >= (long)BB * HW * KPATCH) return;
  int c = (int)(idx % CC);
  int rest = (int)(idx / CC);
  int tap = rest % 9;
  int row = rest / 9;
  int b = row / HW, pos = row % HW;
  int h = pos / WW, w = pos % WW;
  int u = h + tap / 3 - 1, v = w + tap % 3 - 1;
  float out = 0.0f;
  if (u >= 0 && u < HH && v >= 0 && v < WW) {
    int t = targets[(size_t)b * TGTN * TGTN + h * 16 * TGTN + w * 16];
    if (t >= 0 && t < TT) {
      // half-pixel bilinear 12->24; index clamping == jax's renormalized edge weights
      int i0 = (u >> 1) - 1 + (u & 1);
      float f = (u & 1) ? 0.25f : 0.75f;
      int iy0 = i0 < 0 ? 0 : i0;
      int iy1 = (i0 + 1 > 11) ? 11 : (i0 + 1);
      float wy0 = 1.0f - f, wy1 = f;
      int j0 = (v >> 1) - 1 + (v & 1);
      float g = (v & 1) ? 0.25f : 0.75f;
      int jx0 = j0 < 0 ? 0 : j0;
      int jx1 = (j0 + 1 > 11) ? 11 : (j0 + 1);
      float wx0 = 1.0f - g, wx1 = g;
      const float* Yb = Y1 + (size_t)(b * TT + t) * 9 * CC + c;
      out = wy0 * wx0 * Yb[(size_t)(rcat12(iy0) * 3 + rcat12(jx0)) * CC]
          + wy0 * wx1 * Yb[(size_t)(rcat12(iy0) * 3 + rcat12(jx1)) * CC]
          + wy1 * wx0 * Yb[(size_t)(rcat12(iy1) * 3 + rcat12(jx0)) * CC]
          + wy1 * wx1 * Yb[(size_t)(rcat12(iy1) * 3 + rcat12(jx1)) * CC];
    }
  }
  yup[idx] = out;
}

// ---------------- im2col of noise_img (zero-padded 3x3 patches) ----------------
__global__ void k_patch(const float* __restrict__ noise, float* __restrict__ patches) {
  long idx = (long)blockIdx.x * blockDim.x + threadIdx.x;
  if (idx >= (long)BB * HW * KPATCH) return;
  int c = (int)(idx % CC);
  int rest = (int)(idx / CC);
  int tap = rest % 9;
  int row = rest / 9;
  int b = row / HW, pos = row % HW;
  int h = pos / WW, w = pos % WW;
  int u = h + tap / 3 - 1, v = w + tap % 3 - 1;
  float val = 0.0f;
  if (u >= 0 && u < HH && v >= 0 && v < WW)
    val = noise[(size_t)(b * HW + u * WW + v) * CC + c];
  patches[idx] = val;
}

// ---------------- v0/v1/v2 = vis + relu(conv) scatter ----------------
__global__ void k_addvis(const float* __restrict__ Zg, const float* __restrict__ v0in,
                         const float* __restrict__ v1in, const float* __restrict__ v2in,
                         float* __restrict__ out) {
  long idx = (long)blockIdx.x * blockDim.x + threadIdx.x;
  if (idx >= (long)BB * HW * NG) return;
  int n = (int)(idx % NG);
  int row = (int)(idx / NG);
  int b = row / HW, pos = row % HW;
  float z = Zg[idx];
  if (n < 512) {
    size_t i = (size_t)(b * 512 + n) * HW + pos;
    out[O_V0 + i] = v0in[i] + z;
  } else if (n < 768) {
    size_t i = (size_t)(b * 256 + (n - 512)) * HW + pos;
    out[O_V1 + i] = v1in[i] + z;
  } else {
    size_t i = (size_t)(b * 128 + (n - 768)) * HW + pos;
    out[O_V2 + i] = v2in[i] + z;
  }
}

// ---------------- img_out = l2norm(l2norm(img) + noise) over C ----------------
__global__ void k_imgout(const float* __restrict__ imgf, const float* __restrict__ noise,
                         float* __restrict__ out) {
  __shared__ float red[256];
  __shared__ float buf[CC];
  int bp = blockIdx.x;
  int b = bp / HW, pos = bp % HW;
  int tid = threadIdx.x;
  float s = 0.0f;
  for (int c = tid; c < CC; c += 256) {
    float x = imgf[(size_t)(b * CC + c) * HW + pos];
    s += x * x;
  }
  red[tid] = s; __syncthreads();
  for (int k = 128; k > 0; k >>= 1) { if (tid < k) red[tid] += red[tid + k]; __syncthreads(); }
  float d1 = fmaxf(sqrtf(red[0]), 1e-12f);
  __syncthreads();
  s = 0.0f;
  for (int c = tid; c < CC; c += 256) {
    float x = imgf[(size_t)(b * CC + c) * HW + pos] / d1 + noise[(size_t)(b * HW + pos) * CC + c];
    buf[c] = x;
    s += x * x;
  }
  red[tid] = s; __syncthreads();
  for (int k = 128; k > 0; k >>= 1) { if (tid < k) red[tid] += red[tid + k]; __syncthreads(); }
  float d2 = fmaxf(sqrtf(red[0]), 1e-12f);
  for (int c = tid; c < CC; c += 256)
    out[(size_t)(b * CC + c) * HW + pos] = buf[c] / d2;
}

// ---------------- txt_out = l2norm(l2norm(txt,axis=T) + text_noise, axis=T) ----------------
__global__ void k_txtout(const float* __restrict__ tf, const float* __restrict__ tn,
                         float* __restrict__ out) {
  __shared__ float red[64];
  int bc = blockIdx.x;
  int b = bc / CC, c = bc % CC;
  int tid = threadIdx.x;
  float s = 0.0f;
  for (int t = tid; t < TT; t += 64) {
    float x = tf[(size_t)(b * TT + t) * CC + c];
    s += x * x;
  }
  red[tid] = s; __syncthreads();
  for (int k = 32; k > 0; k >>= 1) { if (tid < k) red[tid] += red[tid + k]; __syncthreads(); }
  float d1 = fmaxf(sqrtf(red[0]), 1e-12f);
  __syncthreads();
  s = 0.0f;
  for (int t = tid; t < TT; t += 64) {
    size_t i = (size_t)(b * TT + t) * CC + c;
    float x = tf[i] / d1 + tn[i];
    s += x * x;
  }
  red[tid] = s; __syncthreads();
  for (int k = 32; k > 0; k >>= 1) { if (tid < k) red[tid] += red[tid + k]; __syncthreads(); }
  float d2 = fmaxf(sqrtf(red[0]), 1e-12f);
  for (int t = tid; t < TT; t += 64) {
    size_t i = (size_t)(b * TT + t) * CC + c;
    out[i] = (tf[i] / d1 + tn[i]) / d2;
  }
}

extern "C" void kernel_launch(void* const* d_in, const int* in_sizes, int n_in,
                              void* d_out, int out_size, void* d_ws, size_t ws_size,
                              hipStream_t stream) {
  (void)in_sizes; (void)n_in; (void)out_size; (void)ws_size;
  const float* imgf = (const float*)d_in[0];
  const float* tf   = (const float*)d_in[1];
  const int*   tgt  = (const int*)d_in[2];
  const float* vis0 = (const float*)d_in[3];
  const float* vis1 = (const float*)d_in[4];
  const float* vis2 = (const float*)d_in[5];
  const float* tn   = (const float*)d_in[6];
  const float* w1   = (const float*)d_in[7];
  const float* b1   = (const float*)d_in[8];
  const float* w2   = (const float*)d_in[9];
  const float* b2   = (const float*)d_in[10];
  const float* g0w  = (const float*)d_in[11];
  const float* g0b  = (const float*)d_in[12];
  const float* g1w  = (const float*)d_in[13];
  const float* g1b  = (const float*)d_in[14];
  const float* g2w  = (const float*)d_in[15];
  const float* g2b  = (const float*)d_in[16];

  float* wsf = (float*)d_ws;
  size_t off = 0;
  float* Wsum  = wsf + off; off += (size_t)9 * CC * CC;      // [cat][c][o]
  float* W2c   = wsf + off; off += (size_t)KPATCH * CC;
  float* Gc    = wsf + off; off += (size_t)KPATCH * NG;
  float* gbias = wsf + off; off += NG;
  float* Y1    = wsf + off; off += (size_t)BB * TT * 9 * CC;
  float* yup   = wsf + off; off += (size_t)BB * HW * KPATCH; // reused as patches
  float* noise = wsf + off; off += (size_t)BB * HW * CC;
  float* Zg    = wsf + off; off += (size_t)BB * HW * NG;
  float* gate  = wsf + off; off += 2;
  float* rowGate = wsf + off; off += BB * HW;
  int* cnts = (int*)(wsf + off); off += 4;

  float* out = (float*)d_out;

  // gate logic + loss
  k_count<<<BB, 256, 0, stream>>>(tgt, cnts);
  k_gate<<<1, 1, 0, stream>>>(cnts, gate, out + O_LOSS);
  k_rowgate<<<(BB * HW + 255) / 256, 256, 0, stream>>>(tgt, gate, rowGate);

  // weight restructuring
  k_wsum<<<(9 * CC * CC + 255) / 256, 256, 0, stream>>>(w1, Wsum);
  k_w2c<<<(KPATCH * CC + 255) / 256, 256, 0, stream>>>(w2, W2c);
  k_gc<<<(int)(((long)KPATCH * NG + 255) / 256), 256, 0, stream>>>(g0w, g1w, g2w, g0b, g1b, g2b, Gc, gbias);

  // Stage 1: Y1[bt,cat,o] = relu(X * Wsum[cat] + b1)   (M=342,N=512,K=512, z=cat)
  {
    dim3 grid(CC / 64, (BB * TT + 31) / 32, 9);
    k_wmma_gemm_tiled<<<grid, 256, 0, stream>>>(
        tn, CC, 0,
        Wsum, CC, (long)CC * CC,
        b1, nullptr, 1,
        Y1, 9 * CC, CC,
        BB * TT, CC, CC, 1);
  }

  // Stage 2: build upsampled conv2 im2col, then noise = relu(yup*W2c + b2) * rowGate
  k_yup<<<(int)(((long)BB * HW * KPATCH + 255) / 256), 256, 0, stream>>>(tgt, Y1, yup);
  {
    dim3 grid(CC / 64, (BB * HW) / 32, 1);
    k_wmma_gemm_tiled<<<grid, 256, 0, stream>>>(
        yup, KPATCH, 0,
        W2c, CC, 0,
        b2, rowGate, 1,
        noise, CC, 0,
        BB * HW, CC, KPATCH, 1);
  }

  // Stage 3: g-convs as one GEMM (patches alias yup; stream order makes this safe)
  k_patch<<<(int)(((long)BB * HW * KPATCH + 255) / 256), 256, 0, stream>>>(noise, yup);
  {
    dim3 grid(NG / 64, (BB * HW) / 32, 1);
    k_wmma_gemm_tiled<<<grid, 256, 0, stream>>>(
        yup, KPATCH, 0,
        Gc, NG, 0,
        gbias, nullptr, 1,
        Zg, NG, 0,
        BB * HW, NG, KPATCH, 1);
  }
  k_addvis<<<(int)(((long)BB * HW * NG + 255) / 256), 256, 0, stream>>>(Zg, vis0, vis1, vis2, out);

  // Stage 4: normalizations
  k_imgout<<<BB * HW, 256, 0, stream>>>(imgf, noise, out + O_IMG);
  k_txtout<<<BB * CC, 64, 0, stream>>>(tf, tn, out + O_TXT);
}